// MaskLoss_5738076307627
// MI455X (gfx1250) — compile-verified
//
#include <hip/hip_runtime.h>

// ---------------------------------------------------------------------------
// MaskLoss (label-smoothed BCE-with-logits, reduced to per-batch loss) for
// MI455X / gfx1250.
//
// Shapes: pred, gt : [16, 32, 256, 256] fp32.  out : [16] fp32.
// out[b] = (1/32) * sum_{c,h,w} [ max(x,0) - x*t + log1p(exp(-|x|)) ]
// with t = 0.9*gt + 0.1/256.
//
// Memory-bound: 268 MB streamed once -> ~11.5us floor @ 23.3 TB/s.
// Strategy: B128 non-temporal streaming loads, fp32 math with bare hardware
// v_exp_f32/v_log_f32 (no libm range fixups; log arg is in (1,2]), and an
// exact WMMA-based cross-lane reduction (V_WMMA_F32_16X16X4_F32, ones B).
// ---------------------------------------------------------------------------

typedef float v2f __attribute__((ext_vector_type(2)));
typedef float v4f __attribute__((ext_vector_type(4)));
typedef float v8f __attribute__((ext_vector_type(8)));

#define SMOOTH        0.1f
#define KDIM          256      // last dim (W) used for smoothing constant
#define BATCH         16
#define CCH           32
#define HW            (256 * 256)
#define PER_BATCH     (CCH * HW)                 // 2,097,152 elements
#define BLOCKS_PER_B  256
#define CHUNK         (PER_BATCH / BLOCKS_PER_B) // 8192 elements per block
#define THREADS       256
#define NWAVES        (THREADS / 32)
#define ITERS         (CHUNK / (THREADS * 4))    // 8 float4 iterations

// Stable BCE-with-logits against the smoothed target, using raw hardware
// transcendentals:
//   log1p(exp(-a)) = log2(1 + 2^(-a*log2(e))) * ln(2),  a = |x| >= 0
// The log2 argument is always in (1, 2], the exp2 argument is <= 0, so the
// libm denorm/range fixup sequences are provably unnecessary.
__device__ __forceinline__ float bce_elem(float x, float g) {
    const float LOG2E = 1.4426950408889634f;
    const float LN2   = 0.6931471805599453f;
    float t  = __fmaf_rn(1.0f - SMOOTH, g, SMOOTH / (float)KDIM);
    float ax = __builtin_fabsf(x);
    float e  = __builtin_amdgcn_exp2f(-ax * LOG2E);    // bare v_exp_f32
    float l  = __builtin_amdgcn_logf(1.0f + e) * LN2;  // bare v_log_f32
    return fmaxf(x, 0.0f) - x * t + l;
}

// Exact fp32 sum of the 32 per-lane values using one WMMA.
// A: VGPR0 = p, VGPR1 = 0  ->  A[m][0] = p_m, A[m][2] = p_{m+16}, rest 0.
// B: all ones (4x16).       ->  D[m][n] = p_m + p_{m+16}  (independent of n)
// Lane n<16 holds D[0..7][n] in c[0..7]; lane n>=16 holds D[8..15][n-16].
// Summing a lane's 8 accumulators gives half the total; shfl_xor(16) merges.
__device__ __forceinline__ float wave_reduce_wmma(float p) {
    v2f a; a[0] = p;    a[1] = 0.0f;
    v2f b; b[0] = 1.0f; b[1] = 1.0f;
    v8f c = {};
    c = __builtin_amdgcn_wmma_f32_16x16x4_f32(
            /*neg_a=*/false, a, /*neg_b=*/false, b,
            /*c_mod=*/(short)0, c, /*reuse_a=*/false, /*reuse_b=*/false);
    float s = ((c[0] + c[1]) + (c[2] + c[3])) + ((c[4] + c[5]) + (c[6] + c[7]));
    s += __shfl_xor(s, 16, 32);
    return s;                          // all lanes hold the wave total
}

// Block-level reduction: WMMA wave reduce, then LDS combine across 8 waves.
// Result valid on thread 0. Must be called with all 256 threads active
// (WMMA requires EXEC == all ones).
__device__ __forceinline__ float block_reduce(float p, float* lds) {
    float w    = wave_reduce_wmma(p);
    int   lane = threadIdx.x & 31;
    int   wave = threadIdx.x >> 5;
    if (lane == 0) lds[wave] = w;
    __syncthreads();
    float total = 0.0f;
    if (threadIdx.x == 0) {
#pragma unroll
        for (int i = 0; i < NWAVES; ++i) total += lds[i];
    }
    return total;
}

// Kernel 1: stream one 8192-element chunk per block, emit block partial sums.
__global__ __launch_bounds__(THREADS)
void bce_partial_kernel(const float* __restrict__ pred,
                        const float* __restrict__ gt,
                        float* __restrict__ partials) {
    __shared__ float lds[NWAVES];

    const int  b    = blockIdx.y;
    const long base = (long)b * PER_BATCH + (long)blockIdx.x * CHUNK;
    const v4f* p4   = (const v4f*)(pred + base);
    const v4f* g4   = (const v4f*)(gt + base);
    const int  tid  = threadIdx.x;

    float acc = 0.0f;
#pragma unroll
    for (int it = 0; it < ITERS; ++it) {
        const int i = it * THREADS + tid;
        v4f x = __builtin_nontemporal_load(&p4[i]);   // single-use: NT hint
        v4f g = __builtin_nontemporal_load(&g4[i]);
        acc += bce_elem(x[0], g[0]) + bce_elem(x[1], g[1]);
        acc += bce_elem(x[2], g[2]) + bce_elem(x[3], g[3]);
    }

    float total = block_reduce(acc, lds);
    if (tid == 0) partials[b * BLOCKS_PER_B + blockIdx.x] = total;
}

// Kernel 2: reduce each batch's 256 partials, apply mean over channels.
__global__ __launch_bounds__(THREADS)
void bce_final_kernel(const float* __restrict__ partials,
                      float* __restrict__ out) {
    __shared__ float lds[NWAVES];
    const int b = blockIdx.x;
    float p = partials[b * BLOCKS_PER_B + threadIdx.x];
    float total = block_reduce(p, lds);
    if (threadIdx.x == 0) out[b] = total * (1.0f / (float)CCH);
}

extern "C" void kernel_launch(void* const* d_in, const int* in_sizes, int n_in,
                              void* d_out, int out_size, void* d_ws, size_t ws_size,
                              hipStream_t stream) {
    const float* pred = (const float*)d_in[0];
    const float* gt   = (const float*)d_in[1];
    float*       out  = (float*)d_out;
    float*       ws   = (float*)d_ws;   // needs 16*256*4 = 16 KB

    dim3 grid1(BLOCKS_PER_B, BATCH);
    bce_partial_kernel<<<grid1, THREADS, 0, stream>>>(pred, gt, ws);
    bce_final_kernel<<<BATCH, THREADS, 0, stream>>>(ws, out);
}